// WaveCell_1580547975216
// MI455X (gfx1250) — compile-verified
//
#include <hip/hip_runtime.h>
#include <math.h>

// ---------------- problem constants ----------------
#define NXY      256
#define TSTEPS   128
#define NBATCH   8
// LDS field with halo. Interior cell (i,j), i,j in [-1,256], lives at
//   Y[(i+1)*LSTRIDE + (j+4)]
// LSTRIDE = 268: row stride 16B-aligned (async b128 legal into interior) and
// (ln*268)%64 = (ln*12)%64 distinct for ln=0..15 -> conflict-free
// column-strided fragment reads.
#define LSTRIDE  268
#define LROWS    258
#define LDS_FLOATS (LROWS * LSTRIDE)

typedef __attribute__((ext_vector_type(2))) float v2f;
typedef __attribute__((ext_vector_type(8))) float v8f;

__device__ __forceinline__ int iabs(int x) { return x < 0 ? -x : x; }

// ---------------------------------------------------------------------------
// Precompute: rho = project(lpf(raw_rho)); fuse the step into one float4/cell
//   y = A*y1 - B*y2 + G*lap(y1)
//   A = 8*inv, B = inv*(4-2b), G = inv*c^2, inv = 1/(4+2b)   (dt = 0.5, H = 1)
// Also zeroes the "y2 source" page used for steps t = 0,1.
// ---------------------------------------------------------------------------
__global__ __launch_bounds__(256)
void wave_precompute(const float* __restrict__ raw_rho,
                     const float* __restrict__ bfield,
                     float4* __restrict__ coef,
                     float* __restrict__ zpage) {
    const int i = blockIdx.x;
    const int j = threadIdx.x;

    float s = 0.0f;                       // 3x3 box filter, zero padded
    for (int di = -1; di <= 1; ++di)
        for (int dj = -1; dj <= 1; ++dj) {
            int ii = i + di, jj = j + dj;
            if (ii >= 0 && ii < NXY && jj >= 0 && jj < NXY)
                s += raw_rho[ii * NXY + jj];
        }
    s *= (1.0f / 9.0f);

    const float BETA = 100.0f, ETA = 0.5f, C0v = 1.0f, C1v = 0.9f;
    const float tbe = tanhf(BETA * ETA);
    const float rho = (tbe + tanhf(BETA * (s - ETA))) /
                      (tbe + tanhf(BETA * (1.0f - ETA)));
    const float c   = C0v + (C1v - C0v) * rho;

    const float bv  = bfield[i * NXY + j];
    const float inv = 1.0f / (4.0f + 2.0f * bv);

    float4 o;
    o.x = 8.0f * inv;                 // * y1
    o.y = inv * (4.0f - 2.0f * bv);   // * y2
    o.z = inv * c * c;                // * laplacian
    o.w = 0.0f;
    coef[i * NXY + j] = o;
    zpage[i * NXY + j] = 0.0f;        // y2 source for t = 0,1
}

// ---------------------------------------------------------------------------
// Time stepping: one WG (32 waves) per batch element; y1 field in LDS.
// Each wave owns 8 disjoint 16x16 tiles; the 4-neighbor sum per tile is
//   S*Y + Y*S - 4Y  (S = tridiag(1,0,1))  via 8 chained V_WMMA_F32_16X16X4_F32
// plus one masked cross-tile LDS read per edge cell.
// No register-resident wave state: y2 is read from out[t-2] (L2-hot, written
// two steps ago), and the new field goes LDS <- out[t] via the async copy
// engine. Peak register pressure is ~40 VGPRs -> no scratch spills.
// ---------------------------------------------------------------------------
__global__ __launch_bounds__(1024)
void wave_timestep(const float4* __restrict__ coef,
                   const float* __restrict__ zpage,
                   float* __restrict__ out) {
    extern __shared__ float Y[];

    const int tid  = threadIdx.x;
    const int wave = tid >> 5;
    const int lane = tid & 31;
    const int half = lane >> 4;     // 0: lanes 0-15, 1: lanes 16-31
    const int ln   = lane & 15;     // N coordinate in the C/D fragment
    const int b    = blockIdx.x;

    for (int idx = tid; idx < LDS_FLOATS; idx += 1024) Y[idx] = 0.0f;

    // Constant S16 fragments (tridiagonal 0/1) from lane arithmetic.
    // A (16x4): lanes 0-15 -> M=ln, K=4k+{0,1}; lanes 16-31 -> M=ln, K=4k+{2,3}
    // B (4x16): vgpr0 -> row 4k+half, vgpr1 -> row 4k+2+half, N=ln
    v2f a_sv[4], b_sh[4];
    #pragma unroll
    for (int k = 0; k < 4; ++k) {
        const int ac = 4 * k + 2 * half;
        a_sv[k].x = (iabs(ln - ac)       == 1) ? 1.0f : 0.0f;
        a_sv[k].y = (iabs(ln - (ac + 1)) == 1) ? 1.0f : 0.0f;
        const int br = 4 * k + half;
        b_sh[k].x = (iabs(br - ln)       == 1) ? 1.0f : 0.0f;
        b_sh[k].y = (iabs(br + 2 - ln)   == 1) ? 1.0f : 0.0f;
    }

    const float mtop  = (half == 0) ? 1.0f : 0.0f;  // m==0  -> acc[0], lanes 0-15
    const float mbot  = 1.0f - mtop;                // m==15 -> acc[7], lanes 16-31
    const float medge = (ln == 0 || ln == 15) ? 1.0f : 0.0f;

    __syncthreads();

    float* outb = out + (size_t)b * TSTEPS * NXY * NXY;

    #pragma unroll 1
    for (int t = 0; t < TSTEPS; ++t) {
        float* outt = outb + (size_t)t * NXY * NXY;
        // y2 == field two steps ago == out[t-2] (zero page for t = 0,1).
        // Uniform pointer select, no per-lane branching.
        const float* y2p = (t >= 2) ? (outb + (size_t)(t - 2) * NXY * NXY)
                                    : zpage;

        // ---- phase A: compute y_new per tile, stream to out[t] -------------
        #pragma unroll 2
        for (int s = 0; s < 8; ++s) {
            const int tile = wave + 32 * s;          // 256 tiles / 32 waves
            const int r0 = (tile >> 4) * 16;
            const int c0 = (tile & 15) * 16;
            // masked edge-column address: ln==0 -> col c0-1, ln==15 -> col c0+16
            const int ecol = (ln == 15) ? (c0 + 20) : (c0 + 3);

            // own-cell fragment (C/D layout: M = r + 8*half, N = ln)
            v8f y1f;
            #pragma unroll
            for (int r = 0; r < 8; ++r)
                y1f[r] = Y[(r0 + r + 8 * half + 1) * LSTRIDE + (c0 + ln + 4)];

            v8f acc;
            #pragma unroll
            for (int r = 0; r < 8; ++r) acc[r] = -4.0f * y1f[r];

            // vertical neighbor sum: acc += S * Y
            #pragma unroll
            for (int k = 0; k < 4; ++k) {
                v2f bv;
                bv.x = Y[(r0 + 4 * k + half + 1)     * LSTRIDE + (c0 + ln + 4)];
                bv.y = Y[(r0 + 4 * k + 2 + half + 1) * LSTRIDE + (c0 + ln + 4)];
                acc = __builtin_amdgcn_wmma_f32_16x16x4_f32(
                          false, a_sv[k], false, bv, (short)0, acc, false, false);
            }
            // horizontal neighbor sum: acc += Y * S
            #pragma unroll
            for (int k = 0; k < 4; ++k) {
                v2f av;
                av.x = Y[(r0 + ln + 1) * LSTRIDE + (c0 + 4 * k + 2 * half + 4)];
                av.y = Y[(r0 + ln + 1) * LSTRIDE + (c0 + 4 * k + 2 * half + 5)];
                acc = __builtin_amdgcn_wmma_f32_16x16x4_f32(
                          false, av, false, b_sh[k], (short)0, acc, false, false);
            }

            // branch-free cross-tile corrections (static acc indices)
            acc[0] += mtop * Y[(r0)      * LSTRIDE + (c0 + ln + 4)]; // row r0-1
            acc[7] += mbot * Y[(r0 + 17) * LSTRIDE + (c0 + ln + 4)]; // row r0+16
            #pragma unroll
            for (int r = 0; r < 8; ++r)
                acc[r] += medge * Y[(r0 + r + 8 * half + 1) * LSTRIDE + ecol];

            // pointwise update + streamed store (y2 from L2-hot out[t-2])
            #pragma unroll
            for (int r = 0; r < 8; ++r) {
                const int gidx = (r0 + r + 8 * half) * NXY + (c0 + ln);
                const float4 cf = coef[gidx];
                const float v = cf.x * y1f[r] - cf.y * y2p[gidx] + cf.z * acc[r];
                outt[gidx] = v;
            }
        }

        // all stores coherent before readback; all neighbor reads of old Y
        // done before Y is overwritten
        asm volatile("s_wait_storecnt 0x0" ::: "memory");
        __syncthreads();

        // ---- phase B: async copy out[t] -> LDS (no VGPR round-trip) --------
        // 16384 16B quads, coalesced 512B/wave-instruction on the global side,
        // 16B-aligned on both sides.
        #pragma unroll
        for (int k = 0; k < 16; ++k) {
            const int q = tid + 1024 * k;
            const int qi = q >> 6;           // row 0..255
            const int qc = q & 63;           // column quad
            const unsigned lds_off =
                (unsigned)(uintptr_t)&Y[(qi + 1) * LSTRIDE + 4 * qc + 4];
            const unsigned goff = (unsigned)q * 16u;
            asm volatile("global_load_async_to_lds_b128 %0, %1, %2 offset:0"
                         :: "v"(lds_off), "v"(goff), "s"(outt)
                         : "memory");
        }
        asm volatile("s_wait_asynccnt 0x0" ::: "memory");
        __syncthreads();
    }
}

// ---------------------------------------------------------------------------
extern "C" void kernel_launch(void* const* d_in, const int* in_sizes, int n_in,
                              void* d_out, int out_size, void* d_ws, size_t ws_size,
                              hipStream_t stream) {
    // d_in[0] = x [8,128] (unused by the recurrence),
    // d_in[1] = raw_rho [256,256], d_in[2] = b [256,256]
    const float* raw_rho = (const float*)d_in[1];
    const float* bfield  = (const float*)d_in[2];
    float4* coef = (float4*)d_ws;                                  // 1 MB
    float*  zpage = (float*)((char*)d_ws + NXY * NXY * sizeof(float4)); // 256 KB

    wave_precompute<<<dim3(NXY), dim3(NXY), 0, stream>>>(raw_rho, bfield,
                                                         coef, zpage);

    const size_t lds_bytes = (size_t)LDS_FLOATS * sizeof(float); // ~270 KB/WGP
    wave_timestep<<<dim3(NBATCH), dim3(1024), lds_bytes, stream>>>(
        coef, zpage, (float*)d_out);
}